// ADRiverDynamics_21165598835080
// MI455X (gfx1250) — compile-verified
//
#include <hip/hip_runtime.h>
#include <hip/hip_bf16.h>
#include <math.h>

#define NB   3072
#define CC   64
#define KK   16
#define NPTS 24576          // B*L*N
#define NBL  8              // B*L
#define TAU_INV (1.0f/0.15f)

typedef __attribute__((ext_vector_type(2))) float v2f;
typedef __attribute__((ext_vector_type(8))) float v8f;

// ---------------------------------------------------------------- zero accums
__global__ void k_zero(float* fg_sum, float* bn_sum, float* bn_sumsq) {
    int t = threadIdx.x;
    if (t < NBL * 3) fg_sum[t] = 0.f;
    if (t < CC)      { bn_sum[t] = 0.f; bn_sumsq[t] = 0.f; }
}

// ---------------------------------------------------------------- KNN (K=16)
// one wave (32 lanes) per query point; per-lane register top-16 then
// LDS + shuffle arg-min merge.
__global__ void k_knn(const float* __restrict__ xyz, int* __restrict__ knn) {
    __shared__ float sd[32 * KK];
    __shared__ int   si[32 * KK];
    const int gp   = blockIdx.x;
    const int lane = threadIdx.x;
    const int bl   = gp / NB;
    const int base = bl * NB;
    const float qx = xyz[gp * 3 + 0];
    const float qy = xyz[gp * 3 + 1];
    const float qz = xyz[gp * 3 + 2];

    float dist[KK]; int idv[KK];
#pragma unroll
    for (int t = 0; t < KK; ++t) { dist[t] = 3.4e38f; idv[t] = 0; }

    for (int j = lane; j < NB; j += 32) {
        const float* p = &xyz[(size_t)(base + j) * 3];
        float dx = p[0] - qx, dy = p[1] - qy, dz = p[2] - qz;
        float d2 = dx * dx + dy * dy + dz * dz;
        if (d2 < dist[KK - 1]) {
            dist[KK - 1] = d2; idv[KK - 1] = j;
#pragma unroll
            for (int t = KK - 1; t >= 1; --t) {
                if (dist[t] < dist[t - 1]) {
                    float td = dist[t]; dist[t] = dist[t - 1]; dist[t - 1] = td;
                    int   ti = idv[t];  idv[t]  = idv[t - 1];  idv[t - 1]  = ti;
                }
            }
        }
    }
#pragma unroll
    for (int t = 0; t < KK; ++t) { sd[lane * KK + t] = dist[t]; si[lane * KK + t] = idv[t]; }
    __syncthreads();

    int p = 0;
    for (int r = 0; r < KK; ++r) {
        float v  = sd[lane * KK + p];
        int   id = si[lane * KK + p];
        float bv = v; int bi = id;
#pragma unroll
        for (int s = 16; s >= 1; s >>= 1) {
            float ov = __shfl_xor(bv, s, 32);
            int   oi = __shfl_xor(bi, s, 32);
            if (ov < bv || (ov == bv && oi < bi)) { bv = ov; bi = oi; }
        }
        if (bv == v && bi == id) ++p;        // unique winner advances
        if (lane == 0) knn[gp * KK + r] = bi;
    }
}

// ------------------------------------------------- heads + gate GEMM (WMMA)
__global__ void k_heads_gate(const float* __restrict__ f,
                             const float* __restrict__ Wgate, const float* __restrict__ bgate,
                             const float* __restrict__ Wf, const float* __restrict__ bf,
                             const float* __restrict__ Wd, const float* __restrict__ bd,
                             const float* __restrict__ Wu, const float* __restrict__ bu,
                             float* __restrict__ gate, float* __restrict__ flow,
                             float* __restrict__ diffeff, float* __restrict__ fg_sum) {
    __shared__ float sf[16][65];
    const int tile = blockIdx.x;            // 1536 tiles of 16 points
    const int lane = threadIdx.x;
    const int p0   = tile * 16;

    for (int i = lane; i < 16 * CC; i += 32) {
        int m = i >> 6, c = i & 63;
        sf[m][c] = f[(size_t)(p0 + m) * CC + c];
    }
    __syncthreads();

    const int m = lane & 15, hi = lane >> 4, n = lane & 15;
    v8f acc[4];
#pragma unroll
    for (int nt = 0; nt < 4; ++nt)
#pragma unroll
        for (int r = 0; r < 8; ++r) acc[nt][r] = 0.f;

    for (int kt = 0; kt < 16; ++kt) {
        const int kb = kt * 4;
        v2f a; a.x = sf[m][kb + 2 * hi]; a.y = sf[m][kb + 2 * hi + 1];
#pragma unroll
        for (int nt = 0; nt < 4; ++nt) {
            const int o = nt * 16 + n;
            v2f b; b.x = Wgate[o * CC + kb + 2 * hi];
                   b.y = Wgate[o * CC + kb + 2 * hi + 1];
            acc[nt] = __builtin_amdgcn_wmma_f32_16x16x4_f32(
                false, a, false, b, (short)0, acc[nt], false, false);
        }
    }
#pragma unroll
    for (int nt = 0; nt < 4; ++nt) {
        const int o = nt * 16 + n;
        const float bo = bgate[o];
#pragma unroll
        for (int r = 0; r < 8; ++r) {
            const int mm = r + 8 * hi;
            float v = acc[nt][r] + bo;
            gate[(size_t)(p0 + mm) * CC + o] = 1.f / (1.f + expf(-v));
        }
    }
    // scalar heads on lanes 0..15 (each owns one point)
    if (lane < 16) {
        float a0 = bf[0], a1 = bf[1], a2 = bf[2], ad = bd[0], au = bu[0];
        for (int c = 0; c < CC; ++c) {
            float v = sf[lane][c];
            a0 += v * Wf[0 * CC + c];
            a1 += v * Wf[1 * CC + c];
            a2 += v * Wf[2 * CC + c];
            ad += v * Wd[c];
            au += v * Wu[c];
        }
        const int gp = p0 + lane;
        flow[gp * 3 + 0] = a0; flow[gp * 3 + 1] = a1; flow[gp * 3 + 2] = a2;
        float sp = fmaxf(ad, 0.f) + log1pf(expf(-fabsf(ad)));   // softplus
        float sg = 1.f / (1.f + expf(-au));                     // sigmoid
        diffeff[gp] = sp * (1.f + sg);
        const int bl = gp / NB;
        atomicAdd(&fg_sum[bl * 3 + 0], a0);
        atomicAdd(&fg_sum[bl * 3 + 1], a1);
        atomicAdd(&fg_sum[bl * 3 + 2], a2);
    }
}

// ------------------------------------------------- global-flow MLP (8 waves)
__global__ void k_gatemlp(const float* __restrict__ Wg1, const float* __restrict__ bg1,
                          const float* __restrict__ Wg2, const float* __restrict__ bg2,
                          const float* __restrict__ fg_sum, float* __restrict__ fgf) {
    __shared__ float h[NBL][CC];
    const int tid = threadIdx.x;            // 256 = 8 waves
    const int w = tid >> 5, lane = tid & 31;
    const float inv = 1.f / (float)NB;
    const float g0 = fg_sum[w * 3 + 0] * inv;
    const float g1 = fg_sum[w * 3 + 1] * inv;
    const float g2 = fg_sum[w * 3 + 2] * inv;
    for (int c = lane; c < CC; c += 32) {
        float v = Wg1[c * 3 + 0] * g0 + Wg1[c * 3 + 1] * g1 + Wg1[c * 3 + 2] * g2 + bg1[c];
        h[w][c] = fmaxf(v, 0.f);
    }
    __syncthreads();
    for (int o = lane; o < CC; o += 32) {
        float acc = bg2[o];
        for (int c = 0; c < CC; ++c) acc += h[w][c] * Wg2[o * CC + c];
        fgf[w * CC + o] = acc;
    }
}

// ------------------------------- neighbor aggregation: diffusion + advection
__global__ void k_neigh(const float* __restrict__ f, const float* __restrict__ xyz,
                        const int* __restrict__ knn, const float* __restrict__ flow,
                        const float* __restrict__ diffeff, const float* __restrict__ gate,
                        const float* __restrict__ fgf, float* __restrict__ pre,
                        float* __restrict__ ndist, float* __restrict__ nvar) {
    const int gp   = blockIdx.x;
    const int lane = threadIdx.x;
    const int bl   = gp / NB;
    const int base = bl * NB;

    float fx = flow[gp * 3 + 0], fy = flow[gp * 3 + 1], fz = flow[gp * 3 + 2];
    float fn = sqrtf(fx * fx + fy * fy + fz * fz);
    float fi = 1.f / fmaxf(fn, 1e-6f);
    fx *= fi; fy *= fi; fz *= fi;
    const float qx = xyz[gp * 3 + 0], qy = xyz[gp * 3 + 1], qz = xyz[gp * 3 + 2];

    // lanes 16..31 duplicate lanes 0..15 (16-group shuffles keep halves separate)
    const int k16 = lane & 15;
    const int j   = knn[gp * KK + k16];
    const float* pj = &xyz[(size_t)(base + j) * 3];
    float dx = pj[0] - qx, dy = pj[1] - qy, dz = pj[2] - qz;
    float dn = sqrtf(dx * dx + dy * dy + dz * dz);
    float di = 1.f / fmaxf(dn, 1e-6f);
    float t  = (dx * fx + dy * fy + dz * fz) * di * TAU_INV;

    float mx = t;
#pragma unroll
    for (int s = 8; s >= 1; s >>= 1) mx = fmaxf(mx, __shfl_xor(mx, s, 32));
    float e = expf(t - mx);
    float ss = e;
#pragma unroll
    for (int s = 8; s >= 1; s >>= 1) ss += __shfl_xor(ss, s, 32);
    const float w = e / ss;

    float sd = dn, sq = dn * dn;
#pragma unroll
    for (int s = 8; s >= 1; s >>= 1) { sd += __shfl_xor(sd, s, 32); sq += __shfl_xor(sq, s, 32); }
    const float dmean = sd * (1.f / 16.f);
    const float dvar  = sq * (1.f / 16.f) - dmean * dmean;   // biased variance
    if (lane == 0) { ndist[gp] = dmean; nvar[gp] = dvar; }

    // each lane owns channels (lane, lane+32)
    float am0 = 0.f, am1 = 0.f, af0 = 0.f, af1 = 0.f;
#pragma unroll
    for (int k = 0; k < KK; ++k) {
        const int   jk = __shfl(j, k, 32);
        const float wk = __shfl(w, k, 32);
        const float* fj = &f[(size_t)(base + jk) * CC];
        const float v0 = fj[lane], v1 = fj[lane + 32];
        am0 += v0; am1 += v1; af0 += wk * v0; af1 += wk * v1;
    }
    const float de  = diffeff[gp];
    const float f0  = f[(size_t)gp * CC + lane],       f1  = f[(size_t)gp * CC + lane + 32];
    const float g0  = gate[(size_t)gp * CC + lane],    g1  = gate[(size_t)gp * CC + lane + 32];
    const float fg0 = fgf[bl * CC + lane],             fg1 = fgf[bl * CC + lane + 32];
    pre[(size_t)gp * CC + lane]      = de * (am0 * (1.f / 16.f) - f0) + (af0 - f0) + g0 * fg0;
    pre[(size_t)gp * CC + lane + 32] = de * (am1 * (1.f / 16.f) - f1) + (af1 - f1) + g1 * fg1;
}

// ---------------------------------- reaction part 1: x = rcat*Wr1^T (WMMA)
__global__ void k_reac1(const float* __restrict__ f, const float* __restrict__ flow,
                        const float* __restrict__ ndist, const float* __restrict__ nvar,
                        const float* __restrict__ Wr1, const float* __restrict__ br1,
                        float* __restrict__ xbuf, float* __restrict__ bn_sum,
                        float* __restrict__ bn_sumsq) {
    __shared__ float s[16][73];             // rcat padded 69 -> 72
    const int tile = blockIdx.x, lane = threadIdx.x;
    const int p0 = tile * 16;
    for (int i = lane; i < 16 * 72; i += 32) {
        const int m = i / 72, k = i % 72;
        const int gp = p0 + m;
        float v;
        if (k < 64)      v = f[(size_t)gp * CC + k];
        else if (k < 67) v = flow[gp * 3 + (k - 64)];
        else if (k == 67) v = ndist[gp];
        else if (k == 68) v = nvar[gp];
        else              v = 0.f;
        s[m][k] = v;
    }
    __syncthreads();

    const int m = lane & 15, hi = lane >> 4, n = lane & 15;
    v8f acc[4];
#pragma unroll
    for (int nt = 0; nt < 4; ++nt)
#pragma unroll
        for (int r = 0; r < 8; ++r) acc[nt][r] = 0.f;

    for (int kt = 0; kt < 18; ++kt) {
        const int kb = kt * 4;
        const int k0 = kb + 2 * hi, k1 = k0 + 1;
        v2f a; a.x = s[m][k0]; a.y = s[m][k1];
#pragma unroll
        for (int nt = 0; nt < 4; ++nt) {
            const int o = nt * 16 + n;
            v2f b;
            b.x = (k0 < 69) ? Wr1[o * 69 + k0] : 0.f;
            b.y = (k1 < 69) ? Wr1[o * 69 + k1] : 0.f;
            acc[nt] = __builtin_amdgcn_wmma_f32_16x16x4_f32(
                false, a, false, b, (short)0, acc[nt], false, false);
        }
    }
#pragma unroll
    for (int nt = 0; nt < 4; ++nt) {
        const int o = nt * 16 + n;
        const float bo = br1[o];
        float cs = 0.f, cq = 0.f;
#pragma unroll
        for (int r = 0; r < 8; ++r) {
            float v = acc[nt][r] + bo;
            xbuf[(size_t)(p0 + r + 8 * hi) * CC + o] = v;
            cs += v; cq += v * v;
        }
        cs += __shfl_xor(cs, 16, 32);
        cq += __shfl_xor(cq, 16, 32);
        if (hi == 0) { atomicAdd(&bn_sum[o], cs); atomicAdd(&bn_sumsq[o], cq); }
    }
}

// ------------------------------------------- BN stats -> affine scale/shift
__global__ void k_stats(const float* __restrict__ bn_sum, const float* __restrict__ bn_sumsq,
                        const float* __restrict__ gamma, const float* __restrict__ beta,
                        float* __restrict__ scale, float* __restrict__ shift) {
    const int c = threadIdx.x;              // 64 threads
    const float inv = 1.f / (float)NPTS;
    const float mu  = bn_sum[c] * inv;
    const float var = bn_sumsq[c] * inv - mu * mu;
    const float sc  = gamma[c] * rsqrtf(var + 1e-5f);
    scale[c] = sc;
    shift[c] = beta[c] - mu * sc;
}

// ------------------- reaction part 2 (WMMA) fused with final combination
__global__ void k_reac2(const float* __restrict__ f, const float* __restrict__ xbuf,
                        const float* __restrict__ pre, const float* __restrict__ scale,
                        const float* __restrict__ shift, const float* __restrict__ Wr2,
                        const float* __restrict__ br2, const float* __restrict__ log_dt,
                        float* __restrict__ out) {
    __shared__ float s[16][65];
    const int tile = blockIdx.x, lane = threadIdx.x;
    const int p0 = tile * 16;
    for (int i = lane; i < 16 * CC; i += 32) {
        const int m = i >> 6, c = i & 63;
        float v = xbuf[(size_t)(p0 + m) * CC + c] * scale[c] + shift[c];
        s[m][c] = fmaxf(v, 0.f);            // BN + ReLU
    }
    __syncthreads();

    const int m = lane & 15, hi = lane >> 4, n = lane & 15;
    v8f acc[4];
#pragma unroll
    for (int nt = 0; nt < 4; ++nt)
#pragma unroll
        for (int r = 0; r < 8; ++r) acc[nt][r] = 0.f;

    for (int kt = 0; kt < 16; ++kt) {
        const int kb = kt * 4;
        const int k0 = kb + 2 * hi, k1 = k0 + 1;
        v2f a; a.x = s[m][k0]; a.y = s[m][k1];
#pragma unroll
        for (int nt = 0; nt < 4; ++nt) {
            const int o = nt * 16 + n;
            v2f b; b.x = Wr2[o * CC + k0]; b.y = Wr2[o * CC + k1];
            acc[nt] = __builtin_amdgcn_wmma_f32_16x16x4_f32(
                false, a, false, b, (short)0, acc[nt], false, false);
        }
    }
    float dt = expf(log_dt[0]);
    dt = fminf(fmaxf(dt, 1e-4f), 10.f);
#pragma unroll
    for (int nt = 0; nt < 4; ++nt) {
        const int o = nt * 16 + n;
        const float bo = br2[o];
#pragma unroll
        for (int r = 0; r < 8; ++r) {
            const size_t gp = (size_t)(p0 + r + 8 * hi);
            const float reac = acc[nt][r] + bo;
            out[gp * CC + o] = f[gp * CC + o] + dt * (pre[gp * CC + o] + reac);
        }
    }
}

// ---------------------------------------------------------------- launcher
extern "C" void kernel_launch(void* const* d_in, const int* in_sizes, int n_in,
                              void* d_out, int out_size, void* d_ws, size_t ws_size,
                              hipStream_t stream) {
    const float* f_seq  = (const float*)d_in[0];
    const float* xyz    = (const float*)d_in[1];
    const float* Wf     = (const float*)d_in[2];
    const float* bf     = (const float*)d_in[3];
    const float* Wd     = (const float*)d_in[4];
    const float* bd     = (const float*)d_in[5];
    const float* Wu     = (const float*)d_in[6];
    const float* bu     = (const float*)d_in[7];
    const float* Wg1    = (const float*)d_in[8];
    const float* bg1    = (const float*)d_in[9];
    const float* Wg2    = (const float*)d_in[10];
    const float* bg2    = (const float*)d_in[11];
    const float* Wgate  = (const float*)d_in[12];
    const float* bgate  = (const float*)d_in[13];
    const float* Wr1    = (const float*)d_in[14];
    const float* br1    = (const float*)d_in[15];
    const float* gamma  = (const float*)d_in[16];
    const float* beta   = (const float*)d_in[17];
    const float* Wr2    = (const float*)d_in[18];
    const float* br2    = (const float*)d_in[19];
    const float* log_dt = (const float*)d_in[20];
    float* out = (float*)d_out;

    // workspace layout (floats)
    float* ws = (float*)d_ws;
    int*   knn     = (int*)ws;                      // 24576*16
    float* flow    = ws + NPTS * KK;                // 24576*3
    float* diffeff = flow + NPTS * 3;               // 24576
    float* ndist   = diffeff + NPTS;                // 24576
    float* nvar    = ndist + NPTS;                  // 24576
    float* gate    = nvar + NPTS;                   // 24576*64
    float* pre     = gate + (size_t)NPTS * CC;      // 24576*64
    float* xbuf    = pre  + (size_t)NPTS * CC;      // 24576*64
    float* fg_sum  = xbuf + (size_t)NPTS * CC;      // 24
    float* fgf     = fg_sum + NBL * 3;              // 8*64
    float* bn_sum  = fgf + NBL * CC;                // 64
    float* bn_sumsq= bn_sum + CC;                   // 64
    float* scale   = bn_sumsq + CC;                 // 64
    float* shift   = scale + CC;                    // 64

    const int ntiles = NPTS / 16;                   // 1536

    k_zero<<<1, 256, 0, stream>>>(fg_sum, bn_sum, bn_sumsq);
    k_knn<<<NPTS, 32, 0, stream>>>(xyz, knn);
    k_heads_gate<<<ntiles, 32, 0, stream>>>(f_seq, Wgate, bgate, Wf, bf, Wd, bd, Wu, bu,
                                            gate, flow, diffeff, fg_sum);
    k_gatemlp<<<1, 256, 0, stream>>>(Wg1, bg1, Wg2, bg2, fg_sum, fgf);
    k_neigh<<<NPTS, 32, 0, stream>>>(f_seq, xyz, knn, flow, diffeff, gate, fgf,
                                     pre, ndist, nvar);
    k_reac1<<<ntiles, 32, 0, stream>>>(f_seq, flow, ndist, nvar, Wr1, br1,
                                       xbuf, bn_sum, bn_sumsq);
    k_stats<<<1, 64, 0, stream>>>(bn_sum, bn_sumsq, gamma, beta, scale, shift);
    k_reac2<<<ntiles, 32, 0, stream>>>(f_seq, xbuf, pre, scale, shift, Wr2, br2,
                                       log_dt, out);
}